// GCN_82824149336595
// MI455X (gfx1250) — compile-verified
//
#include <hip/hip_runtime.h>
#include <hip/hip_bf16.h>

typedef __attribute__((ext_vector_type(2))) float v2f;
typedef __attribute__((ext_vector_type(8))) float v8f;

namespace {

constexpr int NNODES = 50000;
constexpr int NEDGES = 800000;
constexpr int F      = 256;
constexpr int NG     = 64;
constexpr int NC     = 10;

__global__ void k_fill(float* __restrict__ p, float v, int n) {
  int i = blockIdx.x * blockDim.x + threadIdx.x;
  if (i < n) p[i] = v;
}

__global__ void k_deg(const int* __restrict__ dst, float* __restrict__ deg, int e) {
  int i = blockIdx.x * blockDim.x + threadIdx.x;
  if (i < e) atomicAdd(&deg[dst[i]], 1.0f);
}

__global__ void k_rsqrt(float* __restrict__ p, int n) {
  int i = blockIdx.x * blockDim.x + threadIdx.x;
  if (i < n) p[i] = rsqrtf(p[i]);
}

// H[16 x 256 tile] = A[row0:row0+16, 0:256] @ W[256, 256]
// 8 waves per block; wave w computes columns [w*32, w*32+32) as two 16x16 tiles.
// f32 WMMA fragment layouts (CDNA5 ISA 7.12.2):
//   A 16x4:  lane m=l&15, VGPR v holds K = 2*(l>>4)+v
//   B 4x16:  lane n=l&15, VGPR v holds K = 2*(l>>4)+v
//   C 16x16: VGPR r holds M = r + 8*(l>>4), N = l&15
__global__ __launch_bounds__(256) void k_gemm(const float* __restrict__ A,
                                              const float* __restrict__ W,
                                              float* __restrict__ H) {
  const int row0 = blockIdx.x * 16;
  const int lane = threadIdx.x & 31;
  const int wave = threadIdx.x >> 5;  // 0..7
  const int col0 = wave * 32;
  const int m    = lane & 15;
  const int kh   = lane >> 4;  // 0 or 1

  const float* arow = A + (size_t)(row0 + m) * F;
  v8f c0 = {};
  v8f c1 = {};
  for (int k0 = 0; k0 < F; k0 += 4) {
    v2f a = *reinterpret_cast<const v2f*>(arow + k0 + 2 * kh);  // 8B aligned
    const float* wp = W + (size_t)(k0 + 2 * kh) * F + col0 + m;
    v2f b0, b1;
    b0.x = wp[0];
    b0.y = wp[F];
    b1.x = wp[16];
    b1.y = wp[F + 16];
    c0 = __builtin_amdgcn_wmma_f32_16x16x4_f32(false, a, false, b0, (short)0, c0,
                                               false, false);
    c1 = __builtin_amdgcn_wmma_f32_16x16x4_f32(false, a, false, b1, (short)0, c1,
                                               false, false);
  }

#pragma unroll
  for (int r = 0; r < 8; ++r) {
    const size_t row = (size_t)(row0 + r + 8 * kh) * F;
    H[row + col0 + m]      = c0[r];
    H[row + col0 + 16 + m] = c1[r];
  }
}

// agg[dst] += h[src] * dinv[src]*dinv[dst]; one 64-thread block per edge, float4 rows.
__global__ __launch_bounds__(64) void k_scatter(const int* __restrict__ src,
                                                const int* __restrict__ dst,
                                                const float* __restrict__ dinv,
                                                const float* __restrict__ H,
                                                float* __restrict__ agg) {
  const int e = blockIdx.x;
  const int s = src[e];
  const int d = dst[e];
  const float coef = dinv[s] * dinv[d];
  const float4 hv =
      reinterpret_cast<const float4*>(H + (size_t)s * F)[threadIdx.x];
  float* op = agg + (size_t)d * F + threadIdx.x * 4;
  atomicAdd(op + 0, hv.x * coef);
  atomicAdd(op + 1, hv.y * coef);
  atomicAdd(op + 2, hv.z * coef);
  atomicAdd(op + 3, hv.w * coef);
}

// act = (relu?)(agg + h*dinv^2 + b), written in place of agg.
__global__ __launch_bounds__(256) void k_finalize(float* __restrict__ agg,
                                                  const float* __restrict__ H,
                                                  const float* __restrict__ dinv,
                                                  const float* __restrict__ bias,
                                                  int relu) {
  const int i = blockIdx.x;
  const int f = threadIdx.x;
  const float di = dinv[i];
  const size_t idx = (size_t)i * F + f;
  float v = agg[idx] + H[idx] * (di * di) + bias[f];
  if (relu) v = fmaxf(v, 0.0f);
  agg[idx] = v;
}

__global__ void k_count(const int* __restrict__ batch, float* __restrict__ cnt,
                        int n) {
  int i = blockIdx.x * blockDim.x + threadIdx.x;
  if (i < n) atomicAdd(&cnt[batch[i]], 1.0f);
}

__global__ __launch_bounds__(64) void k_pool(const int* __restrict__ batch,
                                             const float* __restrict__ act,
                                             float* __restrict__ pool) {
  const int i = blockIdx.x;
  const int g = batch[i];
  const float4 hv =
      reinterpret_cast<const float4*>(act + (size_t)i * F)[threadIdx.x];
  float* op = pool + (size_t)g * F + threadIdx.x * 4;
  atomicAdd(op + 0, hv.x);
  atomicAdd(op + 1, hv.y);
  atomicAdd(op + 2, hv.z);
  atomicAdd(op + 3, hv.w);
}

__global__ __launch_bounds__(64) void k_head(const float* __restrict__ pool,
                                             const float* __restrict__ cnt,
                                             const float* __restrict__ Wl,
                                             const float* __restrict__ bl,
                                             float* __restrict__ out) {
  const int g = blockIdx.x;
  const int c = threadIdx.x;
  if (c >= NC) return;
  float s = 0.f;
  for (int f = 0; f < F; ++f) s += pool[(size_t)g * F + f] * Wl[f * NC + c];
  out[g * NC + c] = s / fmaxf(cnt[g], 1.0f) + bl[c];
}

}  // namespace

extern "C" void kernel_launch(void* const* d_in, const int* in_sizes, int n_in,
                              void* d_out, int out_size, void* d_ws,
                              size_t ws_size, hipStream_t stream) {
  const float* x     = (const float*)d_in[0];
  const int*   ei    = (const int*)d_in[1];
  const int*   batch = (const int*)d_in[2];
  const float* W1    = (const float*)d_in[3];
  const float* b1    = (const float*)d_in[4];
  const float* W2    = (const float*)d_in[5];
  const float* b2    = (const float*)d_in[6];
  const float* W3    = (const float*)d_in[7];
  const float* b3    = (const float*)d_in[8];
  const float* Wl    = (const float*)d_in[9];
  const float* bl    = (const float*)d_in[10];
  float* out = (float*)d_out;

  const int* src = ei;            // edge_index[0]
  const int* dst = ei + NEDGES;   // edge_index[1]

  // Workspace layout (floats): h[N*F] | agg[N*F] | dinv[N] | pool[G*F] | cnt[G]
  float* ws   = (float*)d_ws;
  float* h    = ws;
  float* agg  = h + (size_t)NNODES * F;
  float* dinv = agg + (size_t)NNODES * F;
  float* pool = dinv + NNODES;
  float* cnt  = pool + (size_t)NG * F;

  // Degrees with self-loops -> dinv = rsqrt(deg)
  k_fill<<<(NNODES + 255) / 256, 256, 0, stream>>>(dinv, 1.0f, NNODES);
  k_deg<<<(NEDGES + 255) / 256, 256, 0, stream>>>(dst, dinv, NEDGES);
  k_rsqrt<<<(NNODES + 255) / 256, 256, 0, stream>>>(dinv, NNODES);

  const int gemm_blocks = NNODES / 16;  // 3125, exact
  const float* act_in = x;
  const float* Ws[3] = {W1, W2, W3};
  const float* bs[3] = {b1, b2, b3};
  for (int layer = 0; layer < 3; ++layer) {
    k_gemm<<<gemm_blocks, 256, 0, stream>>>(act_in, Ws[layer], h);
    k_fill<<<(NNODES * F + 255) / 256, 256, 0, stream>>>(agg, 0.0f, NNODES * F);
    k_scatter<<<NEDGES, 64, 0, stream>>>(src, dst, dinv, h, agg);
    k_finalize<<<NNODES, 256, 0, stream>>>(agg, h, dinv, bs[layer],
                                           layer < 2 ? 1 : 0);
    act_in = agg;  // act aliases agg; h is dead for this layer after finalize
  }

  // Global mean pool + linear head. pool and cnt are contiguous -> one fill.
  k_fill<<<(NG * F + NG + 255) / 256, 256, 0, stream>>>(pool, 0.0f, NG * F + NG);
  k_count<<<(NNODES + 255) / 256, 256, 0, stream>>>(batch, cnt, NNODES);
  k_pool<<<NNODES, 64, 0, stream>>>(batch, agg, pool);
  k_head<<<NG, 64, 0, stream>>>(pool, cnt, Wl, bl, out);
}